// CombinedLoss_27900107554853
// MI455X (gfx1250) — compile-verified
//
#include <hip/hip_runtime.h>

typedef float v2f __attribute__((ext_vector_type(2)));
typedef float v4f __attribute__((ext_vector_type(4)));
typedef float v8f __attribute__((ext_vector_type(8)));

constexpr int Bk = 64, Nk = 21, Hk = 128, Wk = 128;
constexpr int TOT     = Bk * Nk * Hk * Wk;   // 22,020,096
constexpr int NV4     = TOT / 4;             // 5,505,024
constexpr int BLOCKS  = 1024;
constexpr int THREADS = 256;                 // 8 wave32 per block
constexpr int NKP     = Bk * Nk;             // 1344 keypoints
// NV4 == BLOCKS*THREADS*21 exactly -> every thread runs 21 full iterations.

__global__ __launch_bounds__(THREADS)
void mse_partial_kernel(const v4f* __restrict__ pred,
                        const v4f* __restrict__ gt,
                        float* __restrict__ partial) {
    const int tid = threadIdx.x;
    unsigned i = blockIdx.x * THREADS + tid;
    float acc = 0.0f;
    // Streaming, single-pass data: non-temporal b128 loads to avoid cache churn.
    #pragma unroll 3
    for (; i < (unsigned)NV4; i += BLOCKS * THREADS) {
        v4f p = __builtin_nontemporal_load(pred + i);
        v4f g = __builtin_nontemporal_load(gt + i);
        v4f d = p - g;
        acc = fmaf(d.x, d.x, acc);
        acc = fmaf(d.y, d.y, acc);
        acc = fmaf(d.z, d.z, acc);
        acc = fmaf(d.w, d.w, acc);
    }

    // Wave32 reduction via V_WMMA_F32_16X16X4_F32 with all-ones B:
    //   A[m][0]=acc(lane m), A[m][2]=acc(lane m+16), A[m][1]=A[m][3]=0
    //   => D[m][n] = acc_m + acc_{m+16} for every n.
    v2f a; a.x = acc;  a.y = 0.0f;
    v2f b; b.x = 1.0f; b.y = 1.0f;   // all-ones B => layout-independent row sums
    v8f c = {};
    c = __builtin_amdgcn_wmma_f32_16x16x4_f32(
        /*neg_a=*/false, a, /*neg_b=*/false, b,
        /*c_mod=*/(short)0, c, /*reuse_a=*/false, /*reuse_b=*/false);

    // Lane 0 holds D[0..7][0] in c[0..7]; lane 16 holds D[8..15][0].
    float s = c[0] + c[1] + c[2] + c[3] + c[4] + c[5] + c[6] + c[7];

    __shared__ float lds[2 * (THREADS / 32)];
    const int lane = tid & 31;
    const int wave = tid >> 5;
    if (lane == 0)  lds[2 * wave + 0] = s;  // sum of lanes 0-7 and 16-23
    if (lane == 16) lds[2 * wave + 1] = s;  // sum of lanes 8-15 and 24-31
    __syncthreads();

    if (tid == 0) {
        float t = 0.0f;
        #pragma unroll
        for (int k = 0; k < 2 * (THREADS / 32); ++k) t += lds[k];
        partial[blockIdx.x] = t;
    }
}

__global__ __launch_bounds__(THREADS)
void finalize_kernel(const float* __restrict__ partial,
                     const float* __restrict__ pred,
                     const float* __restrict__ kp,
                     float* __restrict__ out) {
    __shared__ float s_mse[THREADS];
    __shared__ float s_pck[THREADS];
    const int tid = threadIdx.x;

    // Sum per-block MSE partials (1024 of them).
    float m = 0.0f;
    #pragma unroll
    for (int i = tid; i < BLOCKS; i += THREADS) m += partial[i];

    // Soft-PCK: gather pred at clamped, truncated keypoint coords.
    float p = 0.0f;
    for (int i = tid; i < NKP; i += THREADS) {
        float fx = kp[2 * i + 0];
        float fy = kp[2 * i + 1];
        fx = fminf(fmaxf(fx, 0.0f), (float)(Wk - 1));
        fy = fminf(fmaxf(fy, 0.0f), (float)(Hk - 1));
        const int x = (int)fx;   // truncation matches astype(int32) after clip
        const int y = (int)fy;
        const float v = pred[((long)i * Hk + y) * Wk + x];  // i = b*N + n
        const float d = 1.0f - v;
        p = fmaf(d, d, p);
    }

    s_mse[tid] = m;
    s_pck[tid] = p;
    __syncthreads();
    for (int off = THREADS / 2; off > 0; off >>= 1) {
        if (tid < off) {
            s_mse[tid] += s_mse[tid + off];
            s_pck[tid] += s_pck[tid + off];
        }
        __syncthreads();
    }

    if (tid == 0) {
        const float mse = s_mse[0] * (1.0f / (float)TOT);
        const float pck = s_pck[0] * (1.0f / (float)NKP);
        out[0] = mse + pck;   // total_loss (weights are 1.0)
        out[1] = mse;         // mse_loss
        out[2] = pck;         // soft_pck_loss
    }
}

extern "C" void kernel_launch(void* const* d_in, const int* in_sizes, int n_in,
                              void* d_out, int out_size, void* d_ws, size_t ws_size,
                              hipStream_t stream) {
    const float* pred = (const float*)d_in[0];  // (B,N,H,W) f32
    const float* gt   = (const float*)d_in[1];  // (B,N,H,W) f32
    const float* kp   = (const float*)d_in[2];  // (B,N,2)   f32
    float* out        = (float*)d_out;          // 3 f32 outputs
    float* partials   = (float*)d_ws;           // BLOCKS floats of scratch

    mse_partial_kernel<<<BLOCKS, THREADS, 0, stream>>>(
        (const v4f*)pred, (const v4f*)gt, partials);
    finalize_kernel<<<1, THREADS, 0, stream>>>(partials, pred, kp, out);
}